// GlobalFusion_14310831031049
// MI455X (gfx1250) — compile-verified
//
#include <hip/hip_runtime.h>
#include <hip/hip_bf16.h>
#include <stdint.h>

// ---------------------------------------------------------------------------
// GlobalFusion: sparse-voxel coarse-grid gather + channel concat.
// out[row, 0:64]   = local_features[row]
// out[row, 64:128] = valid ? global_features[grid[flat((coords+lb)//4 - gb)]] : 0
// Pure data movement (~1.04 GB @ 23.3 TB/s). Uses gfx1250 async LDS copies.
// ---------------------------------------------------------------------------

#define C_CH 64
#define GSZ  128          // global_spatial_size (reference constant)
#define SCALE_SHIFT 2     // scale ratio (4,4,4) -> floor-div via >>2

// ---- Kernel 1: fill grid with -1 (B128 stores) ----------------------------
__global__ void gf_init_grid(int4* __restrict__ grid4, int n4) {
    int i = blockIdx.x * blockDim.x + threadIdx.x;
    if (i < n4) grid4[i] = make_int4(-1, -1, -1, -1);
}

// ---- Kernel 2: scatter-max row indices into grid --------------------------
__global__ void gf_scatter(const int* __restrict__ gcoords,
                           int* __restrict__ grid, int n_global) {
    int i = blockIdx.x * blockDim.x + threadIdx.x;
    if (i < n_global) {
        int x = gcoords[3 * i + 0];
        int y = gcoords[3 * i + 1];
        int z = gcoords[3 * i + 2];
        int flat = (x * GSZ + y) * GSZ + z;
        atomicMax(&grid[flat], i);
    }
}

// ---- Kernel 3: one wave32 per row; async LDS staging ----------------------
// lanes 0..15 : local half  (lane*16 bytes of the 256B row)
// lanes 16..31: gathered half
__global__ void gf_fuse(const float* __restrict__ lf,
                        const float* __restrict__ gf,
                        const int*  __restrict__ lc,
                        const int*  __restrict__ grid,
                        const int*  __restrict__ lbase,
                        const int*  __restrict__ gbase,
                        float* __restrict__ out,
                        int n_local) {
    __shared__ __align__(16) float smem[8 * 128];   // 8 waves * 512 B

    const int lane = threadIdx.x & 31;
    const int wave = threadIdx.x >> 5;
    const int wavesPerBlock = blockDim.x >> 5;
    const int gwave  = blockIdx.x * wavesPerBlock + wave;
    const int nwaves = gridDim.x * wavesPerBlock;

    const int lb0 = lbase[0], lb1 = lbase[1], lb2 = lbase[2];
    const int gb0 = gbase[0], gb1 = gbase[1], gb2 = gbase[2];

    // LDS byte address for this lane's 16B slot (flat addr low 32 bits == LDS offset)
    const unsigned lds_off =
        (unsigned)(uintptr_t)(&smem[wave * 128]) + (unsigned)lane * 16u;

    for (int row = gwave; row < n_local; row += nwaves) {
        // previous iteration's async stores must finish reading LDS
        asm volatile("s_wait_asynccnt 0" ::: "memory");

        // --- coarse-grid lookup (wave-uniform; lanes read same addresses) --
        int c0 = lc[3 * row + 0], c1 = lc[3 * row + 1], c2 = lc[3 * row + 2];
        int g0 = ((c0 + lb0) >> SCALE_SHIFT) - gb0;   // arithmetic shift == floor //4
        int g1 = ((c1 + lb1) >> SCALE_SHIFT) - gb1;
        int g2 = ((c2 + lb2) >> SCALE_SHIFT) - gb2;
        bool inb = (g0 >= 0) & (g0 < GSZ) & (g1 >= 0) & (g1 < GSZ) &
                   (g2 >= 0) & (g2 < GSZ);
        int q0 = min(max(g0, 0), GSZ - 1);
        int q1 = min(max(g1, 0), GSZ - 1);
        int q2 = min(max(g2, 0), GSZ - 1);
        int idx = grid[(q0 * GSZ + q1) * GSZ + q2];
        bool valid = inb && (idx >= 0);

        // --- async gather into LDS (no VGPR data movement) -----------------
        if (lane < 16) {
            const float* src = lf + (size_t)row * C_CH + lane * 4;
            asm volatile("global_load_async_to_lds_b128 %0, %1, off"
                         :: "v"(lds_off),
                            "v"((unsigned long long)(uintptr_t)src)
                         : "memory");
        } else if (valid) {
            const float* src = gf + (size_t)idx * C_CH + (lane - 16) * 4;
            asm volatile("global_load_async_to_lds_b128 %0, %1, off"
                         :: "v"(lds_off),
                            "v"((unsigned long long)(uintptr_t)src)
                         : "memory");
        }

        float* dst = out + (size_t)row * (2 * C_CH) +
                     (lane < 16 ? lane * 4 : C_CH + (lane - 16) * 4);

        if (lane >= 16 && !valid) {
            // inactive coarse cell -> zeros, direct B128 store
            *(float4*)dst = make_float4(0.f, 0.f, 0.f, 0.f);
        }

        // gathered data must land in LDS before we stream it out
        asm volatile("s_wait_asynccnt 0" ::: "memory");

        if (lane < 16 || valid) {
            asm volatile("global_store_async_from_lds_b128 %0, %1, off"
                         :: "v"((unsigned long long)(uintptr_t)dst),
                            "v"(lds_off)
                         : "memory");
        }
    }
    asm volatile("s_wait_asynccnt 0" ::: "memory");
}

// ---------------------------------------------------------------------------
extern "C" void kernel_launch(void* const* d_in, const int* in_sizes, int n_in,
                              void* d_out, int out_size, void* d_ws, size_t ws_size,
                              hipStream_t stream) {
    const float* lf = (const float*)d_in[0];   // local_features  (n_local  x 64) f32
    const float* gf = (const float*)d_in[1];   // global_features (n_global x 64) f32
    const int*   lc = (const int*)  d_in[2];   // local_coords    (n_local  x 3)  i32
    const int*   gc = (const int*)  d_in[3];   // global_coords   (n_global x 3)  i32
    const int*   lb = (const int*)  d_in[4];   // local_base  (3) i32
    const int*   gb = (const int*)  d_in[5];   // global_base (3) i32
    // d_in[6]/d_in[7]: spatial sizes (512 / 128) -- fixed reference constants

    const int n_local  = in_sizes[0] / C_CH;
    const int n_global = in_sizes[1] / C_CH;

    int* grid = (int*)d_ws;                    // 128^3 * 4 = 8 MB of scratch
    const int ncell = GSZ * GSZ * GSZ;

    // 1) grid = -1
    {
        int n4 = ncell / 4;
        gf_init_grid<<<(n4 + 255) / 256, 256, 0, stream>>>((int4*)grid, n4);
    }
    // 2) grid[flat(gc)] = max(row)
    gf_scatter<<<(n_global + 255) / 256, 256, 0, stream>>>(gc, grid, n_global);

    // 3) fused gather + concat: one wave per row
    {
        const int threads = 256;                       // 8 waves/block
        const int wavesPerBlock = threads / 32;
        int blocks = (n_local + wavesPerBlock - 1) / wavesPerBlock;
        if (blocks > 131072) blocks = 131072;          // grid-stride beyond this
        gf_fuse<<<blocks, threads, 0, stream>>>(lf, gf, lc, grid, lb, gb,
                                                (float*)d_out, n_local);
    }
}